// GNN_37769942401636
// MI455X (gfx1250) — compile-verified
//
#include <hip/hip_runtime.h>
#include <hip/hip_bf16.h>

typedef __attribute__((ext_vector_type(2))) float v2f;
typedef __attribute__((ext_vector_type(8))) float v8f;

// ---------------------------------------------------------------------------
// Generic GEMM: out[r, c] = sum_k in[r, k] * W[c, k]  (+ bias[c]) (+ PReLU)
// in: [nrows, 128], W: [NCOLS, 128] row-major, out: [nrows, NCOLS]
// Block: (NCOLS/16) waves of 32; each wave computes a 16x16 tile via
// 32 chained V_WMMA_F32_16X16X4_F32 (exact f32 accumulation, RNE).
// ---------------------------------------------------------------------------
template <int NCOLS, bool BIAS, bool PRELU>
__global__ void __launch_bounds__(32 * (NCOLS / 16))
gemm_xwT(const float* __restrict__ in, const float* __restrict__ W,
         const float* __restrict__ bias, const float* __restrict__ alpha_p,
         float* __restrict__ out, int nrows) {
  constexpr int K = 128;
  constexpr int LDA = K + 4;  // pad: conflict-free 64-bit LDS reads (stride 132)
  __shared__ float lds_a[16 * LDA];

  const int tid = threadIdx.x;
  const int wave = tid >> 5;
  const int lane = tid & 31;
  const int row0 = blockIdx.x * 16;
  if (row0 >= nrows) return;  // uniform per block

  // Cooperative load of the 16x128 A tile (float4, fully coalesced).
  for (int i = tid; i < 16 * (K / 4); i += blockDim.x) {
    const int r = i >> 5;          // / (K/4)
    const int c4 = i & 31;         // % (K/4)
    const float4 v = ((const float4*)(in + (size_t)(row0 + r) * K))[c4];
    float* p = &lds_a[r * LDA + c4 * 4];
    p[0] = v.x; p[1] = v.y; p[2] = v.z; p[3] = v.w;
  }
  __syncthreads();

  const int l = lane & 15;
  const int half = lane >> 4;
  const int koff = half * 2;                 // lanes 16-31 carry K+2, K+3
  const int n0 = wave * 16;
  const float* wrow = W + (size_t)(n0 + l) * K;  // B[k][n] = W[n][k]

  v8f c = {};
#pragma unroll
  for (int k = 0; k < K; k += 4) {
    v2f a, b;
    a.x = lds_a[l * LDA + k + koff];
    a.y = lds_a[l * LDA + k + koff + 1];
    b.x = wrow[k + koff];
    b.y = wrow[k + koff + 1];
    c = __builtin_amdgcn_wmma_f32_16x16x4_f32(false, a, false, b,
                                              (short)0, c, false, false);
  }

  const float bv = BIAS ? bias[n0 + l] : 0.0f;
  const float alpha = PRELU ? alpha_p[0] : 0.0f;
#pragma unroll
  for (int v = 0; v < 8; ++v) {
    const int row = row0 + v + 8 * half;     // C/D layout: M = v + 8*half
    float val = c[v] + bv;
    if (PRELU) val = (val >= 0.0f) ? val : alpha * val;
    out[(size_t)row * NCOLS + (n0 + l)] = val;
  }
}

// ---------------------------------------------------------------------------
// Elementwise / scatter helpers
// ---------------------------------------------------------------------------
__global__ void fill_k(float* __restrict__ p, float v, int n) {
  int i = blockIdx.x * blockDim.x + threadIdx.x;
  if (i < n) p[i] = v;
}

__global__ void deg_scatter(const int* __restrict__ dst,
                            const float* __restrict__ w,
                            float* __restrict__ deg, int e) {
  int i = blockIdx.x * blockDim.x + threadIdx.x;
  if (i < e) atomicAdd(&deg[dst[i]], w[i]);
}

__global__ void rsqrt_k(float* __restrict__ p, int n) {
  int i = blockIdx.x * blockDim.x + threadIdx.x;
  if (i < n) p[i] = rsqrtf(p[i]);
}

// agg[n][f] = hW[n][f] * dis[n]^2 + bias[f]   (self loop: dis^2 == 1/deg)
__global__ void agg_init(const float* __restrict__ hW,
                         const float* __restrict__ dis,
                         const float* __restrict__ bias,
                         float* __restrict__ agg, int total) {
  int i = blockIdx.x * blockDim.x + threadIdx.x;
  if (i >= total) return;
  const int node = i >> 7;
  const int f = i & 127;
  const float ds = dis[node];
  agg[i] = hW[i] * (ds * ds) + bias[f];
}

// One wave (32 lanes) per edge; float4 gather + 4 global f32 atomics per lane.
__global__ void edge_scatter(const float* __restrict__ hW,
                             const int* __restrict__ src,
                             const int* __restrict__ dst,
                             const float* __restrict__ w,
                             const float* __restrict__ dis,
                             float* __restrict__ agg, int e) {
  const int wid = (blockIdx.x * blockDim.x + threadIdx.x) >> 5;
  const int lane = threadIdx.x & 31;
  if (wid >= e) return;
  const int s = src[wid];
  const int d = dst[wid];
  const float coef = w[wid] * dis[s] * dis[d];
  const float4 v = ((const float4*)(hW + (size_t)s * 128))[lane];
  float* ap = agg + (size_t)d * 128 + lane * 4;
  atomicAdd(ap + 0, v.x * coef);
  atomicAdd(ap + 1, v.y * coef);
  atomicAdd(ap + 2, v.z * coef);
  atomicAdd(ap + 3, v.w * coef);
}

__global__ void prelu_k(float* __restrict__ p,
                        const float* __restrict__ alpha_p, int n) {
  int i = blockIdx.x * blockDim.x + threadIdx.x;
  if (i >= n) return;
  const float a = alpha_p[0];
  const float v = p[i];
  p[i] = (v >= 0.0f) ? v : a * v;
}

// One wave per node: accumulate feature row into its graph's sum + count.
__global__ void pool_accum(const float* __restrict__ h,
                           const int* __restrict__ batch,
                           float* __restrict__ sums,
                           float* __restrict__ cnt, int n) {
  const int node = (blockIdx.x * blockDim.x + threadIdx.x) >> 5;
  const int lane = threadIdx.x & 31;
  if (node >= n) return;
  const int g = batch[node];
  const float4 v = ((const float4*)(h + (size_t)node * 128))[lane];
  float* sp = sums + (size_t)g * 128 + lane * 4;
  atomicAdd(sp + 0, v.x);
  atomicAdd(sp + 1, v.y);
  atomicAdd(sp + 2, v.z);
  atomicAdd(sp + 3, v.w);
  if (lane == 0) atomicAdd(&cnt[g], 1.0f);
}

// Block (128 threads) per graph: mean + L2 normalize (eps = 1e-12).
__global__ void pool_norm(const float* __restrict__ sums,
                          const float* __restrict__ cnt,
                          float* __restrict__ hg) {
  const int g = blockIdx.x;
  const int t = threadIdx.x;  // 128
  const float c = fmaxf(cnt[g], 1.0f);
  const float v = sums[(size_t)g * 128 + t] / c;
  __shared__ float red[128];
  red[t] = v * v;
  __syncthreads();
  for (int s = 64; s > 0; s >>= 1) {
    if (t < s) red[t] += red[t + s];
    __syncthreads();
  }
  const float denom = fmaxf(sqrtf(red[0]), 1e-12f);
  hg[(size_t)g * 128 + t] = v / denom;
}

// ---------------------------------------------------------------------------
extern "C" void kernel_launch(void* const* d_in, const int* in_sizes, int n_in,
                              void* d_out, int out_size, void* d_ws,
                              size_t ws_size, hipStream_t stream) {
  const float* x         = (const float*)d_in[0];
  const int*   edge_idx  = (const int*)d_in[1];
  const float* edge_attr = (const float*)d_in[2];
  const int*   batch     = (const int*)d_in[3];
  const float* fc1_w = (const float*)d_in[4];
  const float* fc1_b = (const float*)d_in[5];
  const float* gc1_w = (const float*)d_in[6];
  const float* gc1_b = (const float*)d_in[7];
  const float* gc2_w = (const float*)d_in[8];
  const float* gc2_b = (const float*)d_in[9];
  const float* fc2_w = (const float*)d_in[10];
  const float* fc2_b = (const float*)d_in[11];
  const float* a_fc1 = (const float*)d_in[12];
  const float* a_gc1 = (const float*)d_in[13];
  const float* a_gc2 = (const float*)d_in[14];

  const int N = in_sizes[3];        // nodes (batch length)
  const int E = in_sizes[2];        // edges (edge_attr length)
  const int G = out_size / 64;      // graphs
  const int* src = edge_idx;
  const int* dst = edge_idx + E;

  float* ws = (float*)d_ws;
  const size_t nh = (size_t)N * 128;
  float* buf0 = ws;                 // h
  float* buf1 = ws + nh;            // hW
  float* buf2 = ws + 2 * nh;        // agg
  float* dis  = ws + 3 * nh;        // deg -> rsqrt(deg)
  float* sums = dis + N;
  float* cnt  = sums + (size_t)G * 128;
  float* hg   = cnt + G;

  const int NF = N * 128;
  const int T = 256;

  // fc1 + PReLU -> buf0
  gemm_xwT<128, true, true><<<N / 16, 256, 0, stream>>>(x, fc1_w, fc1_b,
                                                        a_fc1, buf0, N);
  // degrees: deg = 1 + segsum(w, dst); dis = rsqrt(deg)
  fill_k<<<(N + T - 1) / T, T, 0, stream>>>(dis, 1.0f, N);
  deg_scatter<<<(E + T - 1) / T, T, 0, stream>>>(dst, edge_attr, dis, E);
  rsqrt_k<<<(N + T - 1) / T, T, 0, stream>>>(dis, N);

  // ---- conv1: buf0 -> buf2 ----
  gemm_xwT<128, false, false><<<N / 16, 256, 0, stream>>>(buf0, gc1_w, nullptr,
                                                          nullptr, buf1, N);
  agg_init<<<(NF + T - 1) / T, T, 0, stream>>>(buf1, dis, gc1_b, buf2, NF);
  edge_scatter<<<(E + 7) / 8, 256, 0, stream>>>(buf1, src, dst, edge_attr, dis,
                                                buf2, E);
  prelu_k<<<(NF + T - 1) / T, T, 0, stream>>>(buf2, a_gc1, NF);

  // ---- conv2: buf2 -> buf0 ----
  gemm_xwT<128, false, false><<<N / 16, 256, 0, stream>>>(buf2, gc2_w, nullptr,
                                                          nullptr, buf1, N);
  agg_init<<<(NF + T - 1) / T, T, 0, stream>>>(buf1, dis, gc2_b, buf0, NF);
  edge_scatter<<<(E + 7) / 8, 256, 0, stream>>>(buf1, src, dst, edge_attr, dis,
                                                buf0, E);
  prelu_k<<<(NF + T - 1) / T, T, 0, stream>>>(buf0, a_gc2, NF);

  // ---- global mean pool + L2 normalize ----
  fill_k<<<(G * 128 + T - 1) / T, T, 0, stream>>>(sums, 0.0f, G * 128);
  fill_k<<<(G + T - 1) / T, T, 0, stream>>>(cnt, 0.0f, G);
  pool_accum<<<(N + 7) / 8, 256, 0, stream>>>(buf0, batch, sums, cnt, N);
  pool_norm<<<G, 128, 0, stream>>>(sums, cnt, hg);

  // ---- fc2 -> d_out [G, 64] ----
  gemm_xwT<64, true, false><<<G / 16, 128, 0, stream>>>(hg, fc2_w, fc2_b,
                                                        nullptr, (float*)d_out,
                                                        G);
}